// GSNN_30124900614661
// MI455X (gfx1250) — compile-verified
//
#include <hip/hip_runtime.h>

typedef float v2f __attribute__((ext_vector_type(2)));
typedef float v8f __attribute__((ext_vector_type(8)));

static __device__ __forceinline__ v8f wmma_f32_k4(v2f a, v2f b, v8f c) {
    // D(16x16,f32) = A(16x4,f32) * B(4x16,f32) + C
    return __builtin_amdgcn_wmma_f32_16x16x4_f32(false, a, false, b, (short)0, c,
                                                 false, false);
}

// ds_swizzle XOR butterfly (group-of-32 mode: [14:10]=xor, [9:5]=or, [4:0]=and)
#define SWZ_X1 0x041f
#define SWZ_X2 0x081f
#define SWZ_X4 0x101f
#define SWZ(v, imm) __int_as_float(__builtin_amdgcn_ds_swizzle(__float_as_int(v), imm))

// ---------------------------------------------------------------- meta prep
__global__ void k_zero_meta(int* p, int n) {
    int i = blockIdx.x * blockDim.x + threadIdx.x;
    if (i < n) p[i] = 0;
}

__global__ void k_count(const int* __restrict__ f_dst, int Ein,
                        const int* __restrict__ f_src, int Eout,
                        const int* __restrict__ dst_idx, int E,
                        int* in_cnt, int* out_cnt, int* deg) {
    int i = blockIdx.x * blockDim.x + threadIdx.x;
    if (i < Ein)  atomicAdd(&in_cnt[f_dst[i]], 1);
    if (i < Eout) atomicAdd(&out_cnt[f_src[i]], 1);
    if (i < E)    atomicAdd(&deg[dst_idx[i]], 1);
}

__global__ void k_scan(const int* __restrict__ in_cnt, const int* __restrict__ out_cnt,
                       int Nf, int* in_off, int* out_off, int* in_cur, int* out_cur) {
    if (threadIdx.x == 0) {
        int a = 0;
        for (int d = 0; d < Nf; ++d) { in_off[d] = a; in_cur[d] = a; a += in_cnt[d]; }
        in_off[Nf] = a;
    }
    if (threadIdx.x == 1) {
        int a = 0;
        for (int d = 0; d < Nf; ++d) { out_off[d] = a; out_cur[d] = a; a += out_cnt[d]; }
        out_off[Nf] = a;
    }
}

// CSR slot = int2{ x: x_edge column, y: weight row (original edge id) }
__global__ void k_fill(const int* __restrict__ f_dst, const int* __restrict__ in_f_edges,
                       int Ein, int* in_cur, int2* ci,
                       const int* __restrict__ f_src, const int* __restrict__ out_f_edges,
                       int Eout, int* out_cur, int2* co) {
    int i = blockIdx.x * blockDim.x + threadIdx.x;
    if (i < Ein) {
        int p = atomicAdd(&in_cur[f_dst[i]], 1);
        ci[p] = make_int2(in_f_edges[i], i);
    }
    if (i < Eout) {
        int p = atomicAdd(&out_cur[f_src[i]], 1);
        co[p] = make_int2(out_f_edges[i], i);
    }
}

static __device__ void seg_sort(int2* a0, int a, int b) {
    for (int i = a + 1; i < b; ++i) {
        int2 v = a0[i];
        int j = i - 1;
        while (j >= a && a0[j].y > v.y) { a0[j + 1] = a0[j]; --j; }
        a0[j + 1] = v;
    }
}

__global__ void k_sortseg(const int* __restrict__ in_off, int2* ci,
                          const int* __restrict__ out_off, int2* co, int Nf) {
    int t = blockIdx.x * blockDim.x + threadIdx.x;
    if (t < Nf)          seg_sort(ci, in_off[t], in_off[t + 1]);
    else if (t < 2 * Nf) { int d = t - Nf; seg_sort(co, out_off[d], out_off[d + 1]); }
}

// ---------------------------------------------------------------- x_edge init
// xeT layout: [E][B]  (batch fastest -> coalesced gathers/scatters per tile)
__global__ void k_init_xe(const float* __restrict__ x, const int* __restrict__ src_idx,
                          float* __restrict__ xeT, int E, int B, int N) {
    int t = blockIdx.x * blockDim.x + threadIdx.x;
    if (t >= E * B) return;
    int b = t % B, e = t / B;
    xeT[e * B + b] = x[b * N + src_idx[e]];
}

// ---------------------------------------------------------------- phase 1
// H_d(16x8) = X_d(16xk) @ W_d(kx8) via f32 WMMA 16x16x4, fused bias+LN+ELU.
// One wave per (batch_tile, function node).  hT layout: [Nf][B][8]
__global__ __launch_bounds__(256) void k_func_nodes(
    const float* __restrict__ xeT, const float* __restrict__ w_in_l,
    const float* __restrict__ b_in_l, const int* __restrict__ in_off,
    const int2* __restrict__ ci, float* __restrict__ hT,
    int B, int Nf, int nwaves) {
    int wave = blockIdx.x * (blockDim.x >> 5) + (threadIdx.x >> 5);
    if (wave >= nwaves) return;
    int lane  = threadIdx.x & 31;
    int btile = wave / Nf;
    int d     = wave - btile * Nf;
    int seg0  = in_off[d];
    int k     = in_off[d + 1] - seg0;

    int half = lane >> 4;            // 0: K=0,1 / D rows 0-7 ; 1: K=2,3 / D rows 8-15
    int l15  = lane & 15;
    int kb   = half << 1;
    int rowA = (btile << 4) + l15;   // A row (batch)
    int ch   = l15;                  // B/D column (channel; 8..15 = pad, never used)
    int ch7  = ch & 7;

    // bias is constant per (node, channel): seed the accumulator with it
    float bias = b_in_l[d * 8 + ch7];
    v8f acc;
#pragma unroll
    for (int g = 0; g < 8; ++g) acc[g] = bias;

    int kfull = k & ~3;
    int base  = seg0 + kb;
#pragma unroll 2
    for (int c0 = 0; c0 < kfull; c0 += 4) {   // steady state: no masks, loads + WMMA
        int2 e0 = ci[base];
        int2 e1 = ci[base + 1];
        v2f A, Bm;
        A.x  = xeT[e0.x * B + rowA];
        A.y  = xeT[e1.x * B + rowA];
        Bm.x = w_in_l[e0.y * 8 + ch7];
        Bm.y = w_in_l[e1.y * 8 + ch7];
        acc  = wmma_f32_k4(A, Bm, acc);
        base += 4;
    }
    if (kfull < k) {                          // wave-uniform tail chunk (1..3 slots)
        int s0 = kfull + kb, s1 = s0 + 1;
        int s0c = min(s0, k - 1), s1c = min(s1, k - 1);
        float m0 = (s0 < k) ? 1.f : 0.f;      // zero A column kills the pad K-slot
        float m1 = (s1 < k) ? 1.f : 0.f;
        int2 e0 = ci[seg0 + s0c];
        int2 e1 = ci[seg0 + s1c];
        v2f A, Bm;
        A.x  = m0 * xeT[e0.x * B + rowA];
        A.y  = m1 * xeT[e1.x * B + rowA];
        Bm.x = w_in_l[e0.y * 8 + ch7];
        Bm.y = w_in_l[e1.y * 8 + ch7];
        acc  = wmma_f32_k4(A, Bm, acc);
    }

    if (ch < 8) {                    // single divergent epilogue region
        int hbase = d * (B * 8);
        int rbase = (btile << 4) + (half << 3);
#pragma unroll
        for (int g = 0; g < 8; ++g) {
            float v = acc[g];
            float s1 = v, s2 = v * v;
            s1 += SWZ(s1, SWZ_X1); s2 += SWZ(s2, SWZ_X1);   // XOR butterfly within octet
            s1 += SWZ(s1, SWZ_X2); s2 += SWZ(s2, SWZ_X2);
            s1 += SWZ(s1, SWZ_X4); s2 += SWZ(s2, SWZ_X4);
            float mu  = s1 * 0.125f;
            float var = s2 * 0.125f - mu * mu;
            float r   = rsqrtf(var + 1e-5f);
            float hn  = (v - mu) * r;
            hn = hn > 0.f ? hn : (__expf(hn) - 1.f);
            hT[hbase + (rbase + g) * 8 + ch] = hn;
        }
    }
}

// ---------------------------------------------------------------- phase 2
// V_s(16 x m) = H_s(16x8) @ Wout_s^T(8 x m), scattered into xeT columns.
// One wave per (batch_tile, source node); edge tiles of 16, K=8 via two WMMAs.
__global__ __launch_bounds__(256) void k_edge_out(
    const float* __restrict__ hT, const float* __restrict__ w_out_l,
    const int* __restrict__ out_off, const int2* __restrict__ co,
    float* __restrict__ xeT, int B, int Nf, int nwaves) {
    int wave = blockIdx.x * (blockDim.x >> 5) + (threadIdx.x >> 5);
    if (wave >= nwaves) return;
    int lane  = threadIdx.x & 31;
    int btile = wave / Nf;
    int s     = wave - btile * Nf;
    int seg0  = out_off[s];
    int m     = out_off[s + 1] - seg0;
    if (m == 0) return;

    int half = lane >> 4;
    int l15  = lane & 15;
    int rowA = (btile << 4) + l15;

    const v2f* hv = (const v2f*)(hT + s * (B * 8) + rowA * 8);  // 32B-aligned
    v2f A0 = hv[half];       // channels kb, kb+1
    v2f A1 = hv[half + 2];   // channels 4+kb, 5+kb
    int rb = (btile << 4) + (half << 3);

    for (int t0 = 0; t0 < m; t0 += 16) {
        int eN  = t0 + l15;                 // B/D column = edge within tile
        int eNc = min(eN, m - 1);           // clamp -> unconditional loads
        int2 e  = co[seg0 + eNc];
        const v2f* wv = (const v2f*)(w_out_l + e.y * 8);
        v2f B0 = wv[half], B1 = wv[half + 2];
        v8f acc = {};
        acc = wmma_f32_k4(A0, B0, acc);
        acc = wmma_f32_k4(A1, B1, acc);
        if (eN < m) {                       // pad columns never stored
            float* p = xeT + e.x * B + rb;
#pragma unroll
            for (int g = 0; g < 8; ++g) p[g] += acc[g];  // unique (row,col): no atomics
        }
    }
}

// ---------------------------------------------------------------- final readout
__global__ void k_final(const float* __restrict__ xeT, const int* __restrict__ out_edges,
                        const int* __restrict__ out_dst, const int* __restrict__ deg,
                        float* __restrict__ out, int B, int Eo, int Nout, int o0,
                        float invL) {
    int t = blockIdx.x * blockDim.x + threadIdx.x;
    if (t >= Eo * B) return;
    int b = t % B, kk = t / B;
    int node = out_dst[kk];
    float dg = (float)max(deg[node], 1);
    out[b * Nout + (node - o0)] = xeT[out_edges[kk] * B + b] * invL * rsqrtf(dg);
}

// ---------------------------------------------------------------- launcher
extern "C" void kernel_launch(void* const* d_in, const int* in_sizes, int n_in,
                              void* d_out, int out_size, void* d_ws, size_t ws_size,
                              hipStream_t stream) {
    const float* x        = (const float*)d_in[0];
    const float* w_in     = (const float*)d_in[1];
    const float* b_in     = (const float*)d_in[2];
    const float* w_out    = (const float*)d_in[3];
    const int* src_idx    = (const int*)d_in[4];
    const int* dst_idx    = (const int*)d_in[5];
    const int* in_f_edges = (const int*)d_in[6];
    const int* f_dst      = (const int*)d_in[7];
    const int* out_f_edges= (const int*)d_in[8];
    const int* f_src      = (const int*)d_in[9];
    const int* out_edges  = (const int*)d_in[10];
    const int* out_dst    = (const int*)d_in[11];

    const int B    = 512, C = 8;
    const int E    = in_sizes[4];
    const int Ein  = in_sizes[6];
    const int Eout = in_sizes[8];
    const int Eo   = in_sizes[10];
    const int Nout = in_sizes[12];
    const int N    = in_sizes[0] / B;
    const int L    = in_sizes[1] / (Ein * C);
    const int Nf   = in_sizes[2] / (L * C);
    const int o0   = N - Nout;

    // workspace carve
    float* xeT = (float*)d_ws;                       // E*B
    float* hT  = xeT + (size_t)E * B;                // Nf*B*C
    int* deg     = (int*)(hT + (size_t)Nf * B * C);  // N
    int* in_cnt  = deg + N;                          // Nf
    int* out_cnt = in_cnt + Nf;                      // Nf
    int* in_off  = out_cnt + Nf;                     // Nf+1
    int* out_off = in_off + Nf + 1;                  // Nf+1
    int* in_cur  = out_off + Nf + 1;                 // Nf
    int* out_cur = in_cur + Nf;                      // Nf
    uintptr_t pa = (uintptr_t)(out_cur + Nf);
    pa = (pa + 15u) & ~(uintptr_t)15;                // 16B-align the int2 arrays
    int2* ci = (int2*)pa;                            // Ein
    int2* co = ci + Ein;                             // Eout

    // CSR build (deterministic: atomic fill then per-segment sort by edge id)
    int zn = N + 2 * Nf;
    k_zero_meta<<<(zn + 255) / 256, 256, 0, stream>>>(deg, zn);
    int mx = E > Ein ? E : Ein; if (Eout > mx) mx = Eout;
    k_count<<<(mx + 255) / 256, 256, 0, stream>>>(f_dst, Ein, f_src, Eout,
                                                  dst_idx, E, in_cnt, out_cnt, deg);
    k_scan<<<1, 64, 0, stream>>>(in_cnt, out_cnt, Nf, in_off, out_off, in_cur, out_cur);
    int mx2 = Ein > Eout ? Ein : Eout;
    k_fill<<<(mx2 + 255) / 256, 256, 0, stream>>>(f_dst, in_f_edges, Ein, in_cur, ci,
                                                  f_src, out_f_edges, Eout, out_cur, co);
    k_sortseg<<<(2 * Nf + 255) / 256, 256, 0, stream>>>(in_off, ci, out_off, co, Nf);

    // x_edge = x[:, src_idx]  (transposed [E][B])
    k_init_xe<<<(E * B + 255) / 256, 256, 0, stream>>>(x, src_idx, xeT, E, B, N);

    const int btiles  = B / 16;
    const int nwaves  = btiles * Nf;
    const int wblocks = (nwaves + 7) / 8;            // 8 waves per 256-thread block
    for (int l = 0; l < L; ++l) {
        k_func_nodes<<<wblocks, 256, 0, stream>>>(
            xeT, w_in + (size_t)l * Ein * C, b_in + (size_t)l * Nf * C,
            in_off, ci, hT, B, Nf, nwaves);
        k_edge_out<<<wblocks, 256, 0, stream>>>(
            hT, w_out + (size_t)l * Eout * C, out_off, co,
            xeT, B, Nf, nwaves);
    }

    k_final<<<(Eo * B + 255) / 256, 256, 0, stream>>>(
        xeT, out_edges, out_dst, deg, (float*)d_out, B, Eo, Nout, o0, 1.0f / (float)L);
}